// TensorProduct_24361054502960
// MI455X (gfx1250) — compile-verified
//
#include <hip/hip_runtime.h>
#include <math.h>

// ---------------------------------------------------------------------------
// e3nn channelwise tensor product, B=16384, 128 channels, 17 CG paths.
//   x1: (B, 1152), x2: (B, 16), w: (B, 2176), out: (B, 9088)
// Memory-bound fp32 (~813 MB @ 23.3 TB/s ~= 35 us). Contraction recast as a
// per-batch GEMM  D(128x71) = A(128x9) x B(9x71)  on V_WMMA_F32_16X16X4_F32,
// with the x2-dependent B matrix built per block in LDS.
// ---------------------------------------------------------------------------

typedef __attribute__((ext_vector_type(2))) float v2f;
typedef __attribute__((ext_vector_type(4))) float v4f;
typedef __attribute__((ext_vector_type(8))) float v8f;
typedef __attribute__((ext_vector_type(4))) int   v4i;

// per-path metadata (original path index k)
__constant__ int c_l1[17]   = {0,0,0,0, 1,1,1,1,1,1, 2,2,2,2,2,2,2};
__constant__ int c_l2[17]   = {0,1,2,3, 0,1,1,2,2,3, 0,1,1,2,2,3,3};
__constant__ int c_l3[17]   = {0,1,2,3, 1,0,2,1,3,2, 2,1,3,0,2,1,3};
__constant__ int c_cgOff[18]= {0,1,10,35,84,93,102,147,192,297,402,427,472,577,602,727,832,1077};

// per-output-column metadata for the K-build (n = 0..70), padded to 80
__constant__ int c_kOfN[80] = {
  0,5,13,
  1,1,1, 4,4,4, 7,7,7, 11,11,11, 15,15,15,
  2,2,2,2,2, 6,6,6,6,6, 9,9,9,9,9, 10,10,10,10,10, 14,14,14,14,14,
  3,3,3,3,3,3,3, 8,8,8,8,8,8,8, 12,12,12,12,12,12,12, 16,16,16,16,16,16,16,
  0,0,0,0,0,0,0,0,0};
__constant__ int c_cOfN[80] = {
  0,0,0,
  0,1,2, 0,1,2, 0,1,2, 0,1,2, 0,1,2,
  0,1,2,3,4, 0,1,2,3,4, 0,1,2,3,4, 0,1,2,3,4, 0,1,2,3,4,
  0,1,2,3,4,5,6, 0,1,2,3,4,5,6, 0,1,2,3,4,5,6, 0,1,2,3,4,5,6,
  0,0,0,0,0,0,0,0,0};

// epilogue metadata packed: k | c<<5 | d<<8 | nb<<16   (one b32 load per tile)
#define M(k,c,d,nb) ((k) | ((c)<<5) | ((d)<<8) | ((nb)<<16))
__constant__ int c_metaOfN[80] = {
  M(0,0,1,0), M(5,0,1,1), M(13,0,1,2),
  M(1,0,3,3),  M(1,1,3,3),  M(1,2,3,3),
  M(4,0,3,6),  M(4,1,3,6),  M(4,2,3,6),
  M(7,0,3,9),  M(7,1,3,9),  M(7,2,3,9),
  M(11,0,3,12),M(11,1,3,12),M(11,2,3,12),
  M(15,0,3,15),M(15,1,3,15),M(15,2,3,15),
  M(2,0,5,18), M(2,1,5,18), M(2,2,5,18), M(2,3,5,18), M(2,4,5,18),
  M(6,0,5,23), M(6,1,5,23), M(6,2,5,23), M(6,3,5,23), M(6,4,5,23),
  M(9,0,5,28), M(9,1,5,28), M(9,2,5,28), M(9,3,5,28), M(9,4,5,28),
  M(10,0,5,33),M(10,1,5,33),M(10,2,5,33),M(10,3,5,33),M(10,4,5,33),
  M(14,0,5,38),M(14,1,5,38),M(14,2,5,38),M(14,3,5,38),M(14,4,5,38),
  M(3,0,7,43), M(3,1,7,43), M(3,2,7,43), M(3,3,7,43), M(3,4,7,43), M(3,5,7,43), M(3,6,7,43),
  M(8,0,7,50), M(8,1,7,50), M(8,2,7,50), M(8,3,7,50), M(8,4,7,50), M(8,5,7,50), M(8,6,7,50),
  M(12,0,7,57),M(12,1,7,57),M(12,2,7,57),M(12,3,7,57),M(12,4,7,57),M(12,5,7,57),M(12,6,7,57),
  M(16,0,7,64),M(16,1,7,64),M(16,2,7,64),M(16,3,7,64),M(16,4,7,64),M(16,5,7,64),M(16,6,7,64),
  M(0,0,1,0),M(0,0,1,0),M(0,0,1,0),M(0,0,1,0),M(0,0,1,0),M(0,0,1,0),M(0,0,1,0),M(0,0,1,0),M(0,0,1,0)};
#undef M

// ---------------------------------------------------------------------------
// Setup kernel: 1077 real-CG coefficients (e3nn convention) into ws.
// Racah su2_cg -> complex->real basis change with (-i)^l phasing -> re/im by
// parity. Normalization: reference divides by ||C||_F = sqrt(2l3+1) and later
// multiplies alpha = sqrt((2l3+1)/128); folded together this is a uniform
// 1/sqrt(128), so the epilogue needs no per-path alpha.
// ---------------------------------------------------------------------------

__device__ inline int imax2(int a, int b) { return a > b ? a : b; }
__device__ inline int imin2(int a, int b) { return a < b ? a : b; }

__device__ double dfact(int n) { double r = 1.0; for (int t = 2; t <= n; ++t) r *= (double)t; return r; }

__device__ double su2_cg(int j1, int m1, int j2, int m2, int j3, int m3) {
  if (m1 + m2 != m3) return 0.0;
  double pref = sqrt((2.0 * j3 + 1.0) * dfact(j1 + j2 - j3) * dfact(j1 - j2 + j3) *
                     dfact(-j1 + j2 + j3) / dfact(j1 + j2 + j3 + 1));
  pref *= sqrt(dfact(j1 + m1) * dfact(j1 - m1) * dfact(j2 + m2) * dfact(j2 - m2) *
               dfact(j3 + m3) * dfact(j3 - m3));
  int lo = imax2(0, imax2(j2 - j3 - m1, j1 - j3 + m2));
  int hi = imin2(j1 + j2 - j3, imin2(j1 - m1, j2 + m2));
  double s = 0.0;
  for (int k = lo; k <= hi; ++k) {
    double d = dfact(k) * dfact(j1 + j2 - j3 - k) * dfact(j1 - m1 - k) *
               dfact(j2 + m2 - k) * dfact(j3 - j2 + m1 + k) * dfact(j3 - j1 - m2 + k);
    s += ((k & 1) ? -1.0 : 1.0) / d;
  }
  return pref * s;
}

struct QE { int col; double re, im; };

__device__ int rowq(int l, int ri, QE* o) {
  const double s = 0.70710678118654752440;
  int m = ri - l;
  int cnt;
  if (m < 0) {
    o[0].col = l - m; o[0].re = s;   o[0].im = 0.0;
    o[1].col = l + m; o[1].re = 0.0; o[1].im = -s;
    cnt = 2;
  } else if (m == 0) {
    o[0].col = l; o[0].re = 1.0; o[0].im = 0.0;
    cnt = 1;
  } else {
    double sg = (m & 1) ? -1.0 : 1.0;
    o[0].col = l + m; o[0].re = sg * s; o[0].im = 0.0;
    o[1].col = l - m; o[1].re = 0.0;   o[1].im = sg * s;
    cnt = 2;
  }
  double pr, pi;                       // multiply by (-i)^l
  switch (l & 3) {
    case 0: pr = 1.0;  pi = 0.0;  break;
    case 1: pr = 0.0;  pi = -1.0; break;
    case 2: pr = -1.0; pi = 0.0;  break;
    default: pr = 0.0; pi = 1.0;  break;
  }
  for (int t = 0; t < cnt; ++t) {
    double re = o[t].re * pr - o[t].im * pi;
    double im = o[t].re * pi + o[t].im * pr;
    o[t].re = re; o[t].im = im;
  }
  return cnt;
}

__global__ void cg_setup_kernel(float* __restrict__ cg) {
  int e = blockIdx.x * blockDim.x + threadIdx.x;
  if (e >= 1077) return;
  int k = 16;
  for (int t = 0; t < 17; ++t) { if (e < c_cgOff[t + 1]) { k = t; break; } }
  int l1 = c_l1[k], l2 = c_l2[k], l3 = c_l3[k];
  int n2 = 2 * l2 + 1, n3 = 2 * l3 + 1;
  int local = e - c_cgOff[k];
  int cc = local % n3;
  int bb = (local / n3) % n2;
  int aa = local / (n3 * n2);

  QE qa[2], qb[2], qc[2];
  int na = rowq(l1, aa, qa);
  int nb = rowq(l2, bb, qb);
  int nc = rowq(l3, cc, qc);
  double accre = 0.0, accim = 0.0;
  for (int ia = 0; ia < na; ++ia) {
    double ar = qa[ia].re, ai = -qa[ia].im;           // conj(q1)
    int m1 = qa[ia].col - l1;
    for (int ib = 0; ib < nb; ++ib) {
      double br = qb[ib].re, bi = -qb[ib].im;         // conj(q2)
      int m2 = qb[ib].col - l2;
      double t1r = ar * br - ai * bi;
      double t1i = ar * bi + ai * br;
      for (int ic = 0; ic < nc; ++ic) {
        int m3 = qc[ic].col - l3;
        double coef = su2_cg(l1, m1, l2, m2, l3, m3);
        if (coef == 0.0) continue;
        double cr = qc[ic].re, ci = qc[ic].im;        // q3
        accre += coef * (t1r * cr - t1i * ci);
        accim += coef * (t1r * ci + t1i * cr);
      }
    }
  }
  double val = ((l1 + l2 + l3) & 1) ? accim : accre;
  cg[e] = (float)(val * 0.088388347648318447);        // 1/sqrt(128): ||C|| * alpha folded
}

// ---------------------------------------------------------------------------
// Address-space helpers for async global->LDS. The builtin expects
// (int4 global-AS*, int4 LDS-AS*, imm offset, imm cpol). Generic LDS address
// truncates to the 32-bit LDS offset per the ISA aperture rule; const is
// stripped via the integer round-trip.
// ---------------------------------------------------------------------------
#define __AS1 __attribute__((address_space(1)))
#define __AS3 __attribute__((address_space(3)))
typedef __AS1 v4i* gv4i_p;
typedef __AS3 v4i* sv4i_p;
__device__ __forceinline__ gv4i_p to_as1(const void* p) {
  return (gv4i_p)(unsigned long long)p;
}
__device__ __forceinline__ sv4i_p to_as3(void* p) {
  return (sv4i_p)(unsigned int)(unsigned long long)p;
}

__device__ inline float loadA(const float* __restrict__ x1row, int u, int a) {
  if (a == 0) return x1row[u];
  if (a < 4)  return x1row[128 + u * 3 + (a - 1)];
  if (a < 9)  return x1row[512 + u * 5 + (a - 4)];
  return 0.0f;
}

// ---------------------------------------------------------------------------
// Main kernel: one block per batch row; 256 threads = 8 wave32 = 8 M-tiles.
// ---------------------------------------------------------------------------
__global__ __launch_bounds__(256) void tp_kernel(const float* __restrict__ x1,
                                                 const float* __restrict__ x2,
                                                 const float* __restrict__ w,
                                                 float* __restrict__ out,
                                                 const float* __restrict__ cg) {
  __shared__ float cgs[1088];   // CG table staged in LDS (async)
  __shared__ float x2s[16];
  __shared__ float Klds[960];   // [80][12] B operand, K-major inner

  const int b = blockIdx.x;
  const int tid = threadIdx.x;
  const float* x1row = x1 + (long)b * 1152;
  const float* x2row = x2 + (long)b * 16;

  // ---- Stage CG (4308 B) and x2 (64 B) into LDS ----
#if __has_builtin(__builtin_amdgcn_global_load_async_to_lds_b128)
  {
    // 269 full 16B chunks cover cg[0..1075]; last float scalar-copied (no OOB).
    for (int t = tid; t < 269; t += 256)
      __builtin_amdgcn_global_load_async_to_lds_b128(
          to_as1(cg + t * 4), to_as3(&cgs[t * 4]), 0, 0);
    if (tid < 4)
      __builtin_amdgcn_global_load_async_to_lds_b128(
          to_as1(x2row + tid * 4), to_as3(&x2s[tid * 4]), 0, 0);
    if (tid == 0) cgs[1076] = cg[1076];
#if __has_builtin(__builtin_amdgcn_s_wait_asynccnt)
    __builtin_amdgcn_s_wait_asynccnt(0);
#else
    asm volatile("s_wait_asynccnt 0x0" ::: "memory");
#endif
  }
#else
  for (int t = tid; t < 1077; t += 256) cgs[t] = cg[t];
  if (tid < 16) x2s[tid] = x2row[tid];
#endif
  __syncthreads();

  // ---- Phase 1: K[n][a] = sum_j CG_k[a_local, j, c(n)] * x2[block(j)] ----
  for (int idx = tid; idx < 960; idx += 256) {
    int n = idx / 12, a = idx % 12;
    float v = 0.0f;
    if (n < 71 && a < 9) {
      int k  = c_kOfN[n];
      int c  = c_cOfN[n];
      int l1 = c_l1[k];
      int as = l1 * l1;                 // a-offset of irrep block i: 0,1,4
      int ac = 2 * l1 + 1;
      if (a >= as && a < as + ac) {
        int al = a - as;
        int l2 = c_l2[k], l3 = c_l3[k];
        int n2 = 2 * l2 + 1, d = 2 * l3 + 1;
        const float* cgp = &cgs[c_cgOff[k] + al * n2 * d + c];
        const float* xp  = &x2s[l2 * l2];      // OFF2[j] = l2^2
        for (int b2 = 0; b2 < n2; ++b2) v += cgp[b2 * d] * xp[b2];
      }
    }
    Klds[idx] = v;
  }
  __syncthreads();

  const int lane = tid & 31;
  const int wv   = tid >> 5;    // M-tile 0..7
  const int hi   = lane >> 4;   // half-wave: K pair select
  const int m    = lane & 15;

  // ---- A operand: x1 in 16x4 f32 WMMA layout (3 K-chunks) ----
  v2f Aop[3];
#pragma unroll
  for (int q = 0; q < 3; ++q) {
    int a0 = q * 4 + hi * 2;
    int u  = wv * 16 + m;
    Aop[q].x = loadA(x1row, u, a0);
    Aop[q].y = loadA(x1row, u, a0 + 1);
  }

  v8f acc[5];

#if __has_builtin(__builtin_amdgcn_wmma_f32_16x16x4_f32)
  {
    v8f z = {0.f,0.f,0.f,0.f,0.f,0.f,0.f,0.f};
#pragma unroll
    for (int t = 0; t < 5; ++t) acc[t] = z;
#pragma unroll
    for (int q = 0; q < 3; ++q) {
#pragma unroll
      for (int nt = 0; nt < 5; ++nt) {
        int col = nt * 16 + m;
        v2f Bop = *(const v2f*)&Klds[col * 12 + q * 4 + hi * 2];  // 8B aligned
        acc[nt] = __builtin_amdgcn_wmma_f32_16x16x4_f32(
            false, Aop[q], false, Bop, (short)0, acc[nt], false, false);
      }
    }
  }
#else
  // VALU fallback (host pass / builtin-mismatch): direct dot products.
#pragma unroll
  for (int nt = 0; nt < 5; ++nt) {
#pragma unroll
    for (int r = 0; r < 8; ++r) {
      int uu = wv * 16 + hi * 8 + r;
      int n  = nt * 16 + m;
      float s = 0.0f;
      for (int a = 0; a < 9; ++a) s += loadA(x1row, uu, a) * Klds[n * 12 + a];
      acc[nt][r] = s;
    }
  }
#endif

  // ---- Phase 3: epilogue — w[b,u,k] * D, scatter to e3nn output layout ----
  const int ubase = wv * 16 + hi * 8;             // C/D rows: lanes16-31 -> M+8
#pragma unroll
  for (int nt = 0; nt < 5; ++nt) {
    int n = nt * 16 + m;
    if (n < 71) {
      int meta = c_metaOfN[n];
      int k  = meta & 31;
      int c  = (meta >> 5) & 7;
      int d  = (meta >> 8) & 7;
      int nb = meta >> 16;
      const float* wrow = w + (long)b * 2176 + k * 128 + ubase;   // 32B aligned
      v4f w0 = *(const v4f*)&wrow[0];
      v4f w1 = *(const v4f*)&wrow[4];
      float* orow = out + (long)b * 9088 + nb * 128 + c + (long)ubase * d;
#pragma unroll
      for (int r = 0; r < 4; ++r) orow[(long)r * d] = acc[nt][r] * w0[r];
#pragma unroll
      for (int r = 0; r < 4; ++r) orow[(long)(r + 4) * d] = acc[nt][r + 4] * w1[r];
    }
  }
}

// ---------------------------------------------------------------------------
extern "C" void kernel_launch(void* const* d_in, const int* in_sizes, int n_in,
                              void* d_out, int out_size, void* d_ws, size_t ws_size,
                              hipStream_t stream) {
  const float* x1 = (const float*)d_in[0];
  const float* x2 = (const float*)d_in[1];
  const float* wt = (const float*)d_in[2];
  float* out = (float*)d_out;
  float* cg  = (float*)d_ws;                      // 1077 floats of (alpha-scaled) CG
  int B = in_sizes[0] / 1152;

  cg_setup_kernel<<<(1077 + 255) / 256, 256, 0, stream>>>(cg);
  tp_kernel<<<B, 256, 0, stream>>>(x1, x2, wt, out, cg);
}